// MomentumLIF_523986010137
// MI455X (gfx1250) — compile-verified
//
#include <hip/hip_runtime.h>
#include <stdint.h>

// ---------------- problem constants (from reference) ----------------
#define LIF_N 64
#define LIF_T 64
#define LIF_D 8192
#define LIF_TAU 2.0f
#define LIF_TH 1.0f

// ---------------- tiling ----------------
#define DBLK     1024           // floats of D handled per block per timestep
#define NTHREADS 256            // 8 wave32s; each thread owns 4 floats
#define PIPE     4              // async pipeline depth (power of two)

typedef float v4f __attribute__((ext_vector_type(4)));
typedef int   v4i __attribute__((vector_size(16)));

#ifndef __has_builtin
#define __has_builtin(x) 0
#endif

#if __has_builtin(__builtin_amdgcn_global_load_async_to_lds_b128)
#define MLIF_ASYNC 1
#else
#define MLIF_ASYNC 0
#endif

#if __has_builtin(__builtin_amdgcn_s_wait_asynccnt)
#define MLIF_WAIT(n) __builtin_amdgcn_s_wait_asynccnt(n)
#elif MLIF_ASYNC
#define MLIF_WAIT(n) asm volatile("s_wait_asynccnt " #n ::: "memory")
#else
#define MLIF_WAIT(n) do {} while (0)
#endif

// 16B-per-lane async copy global -> LDS (512B contiguous per wave32,
// tracked by ASYNCcnt). Fallback is a synchronous copy (still correct:
// each wave consumes only the LDS bytes it produced).
__device__ __forceinline__ void async_copy16(const float* __restrict__ gsrc,
                                             v4f* ldst) {
#if MLIF_ASYNC
    __builtin_amdgcn_global_load_async_to_lds_b128(
        (__attribute__((address_space(1))) v4i*)gsrc,
        (__attribute__((address_space(3))) v4i*)ldst,
        /*imm offset*/ 0, /*cpol*/ 0);
#else
    *ldst = *(const v4f*)gsrc;
#endif
}

__global__ __launch_bounds__(NTHREADS)
void MomentumLIF_kernel(const float* __restrict__ x,
                        const float* __restrict__ momentum,
                        const float* __restrict__ lamb,
                        float* __restrict__ out) {
    // typed v4f tile: direct indexing keeps AS(3) inference -> ds_load_b128
    __shared__ v4f tile[PIPE][DBLK / 4];   // 4 * 4KB = 16KB LDS

    const int tid = threadIdx.x;
    const int blocks_per_n = LIF_D / DBLK;            // 8
    const int n   = blockIdx.x / blocks_per_n;        // 0..63
    const int db  = (blockIdx.x % blocks_per_n) * DBLK;
    const int col = db + tid * 4;                      // 16B-aligned lane base

    const float mom   = momentum[0];
    const float lb    = lamb[0];
    const float decay = 1.0f - 1.0f / LIF_TAU;         // 0.5

    const float* xbase = x   + (size_t)n * LIF_T * LIF_D + col;
    float*       obase = out + (size_t)n * LIF_T * LIF_D + col;

    // ---- prologue: fill the pipeline (t = 0..PIPE-1) ----
#pragma unroll
    for (int p = 0; p < PIPE; ++p)
        async_copy16(xbase + (size_t)p * LIF_D, &tile[p][tid]);

    v4f u = {0.f, 0.f, 0.f, 0.f};
    v4f v = {0.f, 0.f, 0.f, 0.f};

#pragma unroll PIPE
    for (int t = 0; t < LIF_T; ++t) {
        const int buf = t & (PIPE - 1);   // compile-time constant per unrolled body

        // async loads complete in order: once ASYNCcnt <= PIPE-1 the oldest
        // outstanding load (buffer `buf`) has landed in LDS.
        MLIF_WAIT(3);

        v4f xt = tile[buf][tid];          // ds_load_b128
        v4f s;

        // v = mom*v + x - lamb*u ; u = decay*u + v ; spike + hard reset
#define LIF_STEP(c)                                            \
        v.c = mom * v.c + xt.c - lb * u.c;                     \
        u.c = decay * u.c + v.c;                               \
        s.c = (u.c >= LIF_TH) ? 1.0f : 0.0f;                   \
        u.c = (u.c >= LIF_TH) ? 0.0f : u.c;
        LIF_STEP(x) LIF_STEP(y) LIF_STEP(z) LIF_STEP(w)
#undef LIF_STEP

        // spikes are write-once, never re-read: non-temporal 128-bit store
        __builtin_nontemporal_store(s, (v4f*)(obase + (size_t)t * LIF_D));

        // xt was consumed above (DScnt wait already executed in-order),
        // so re-using buffer `buf` for t+PIPE is WAR-safe.
        const int tn = t + PIPE;
        if (tn < LIF_T)
            async_copy16(xbase + (size_t)tn * LIF_D, &tile[buf][tid]);
    }
}

extern "C" void kernel_launch(void* const* d_in, const int* in_sizes, int n_in,
                              void* d_out, int out_size, void* d_ws, size_t ws_size,
                              hipStream_t stream) {
    (void)in_sizes; (void)n_in; (void)out_size; (void)d_ws; (void)ws_size;
    const float* x        = (const float*)d_in[0];
    const float* momentum = (const float*)d_in[1];
    const float* lamb     = (const float*)d_in[2];
    float*       out      = (float*)d_out;

    const int grid = LIF_N * (LIF_D / DBLK);   // 64 * 8 = 512 blocks
    MomentumLIF_kernel<<<grid, NTHREADS, 0, stream>>>(x, momentum, lamb, out);
}